// SAGELayer_154618823170
// MI455X (gfx1250) — compile-verified
//
#include <hip/hip_runtime.h>

typedef float v2f __attribute__((ext_vector_type(2)));
typedef float v8f __attribute__((ext_vector_type(8)));

#define FEAT 64
#define WSTRIDE 132   // padded LDS row stride (floats) for W: 132 % 64 = 4 -> no bank conflicts

// ---------------------------------------------------------------------------
// Kernel 1: deterministic CSR-style aggregation.
// src[] is non-decreasing (np.unique-sorted codes), so each node's edges are a
// contiguous range. Group of 64 threads (2 waves) per node, one thread per
// feature. No atomics, no init pass: every (node, feature) written exactly once.
// ---------------------------------------------------------------------------
__global__ __launch_bounds__(256) void sage_agg_kernel(
    const float* __restrict__ x,
    const int* __restrict__ src,
    const int* __restrict__ dst,
    float* __restrict__ agg,
    float* __restrict__ deg,
    int nNodes, int nEdges)
{
    int gt = blockIdx.x * blockDim.x + threadIdx.x;
    int v = gt >> 6;          // node (uniform across each wave: 32 lanes share v)
    int f = gt & 63;          // feature
    if (v >= nNodes) return;

    // lower_bound(v)
    int lo = 0, hi = nEdges;
    while (lo < hi) { int mid = (lo + hi) >> 1; if (src[mid] < v) lo = mid + 1; else hi = mid; }
    int beg = lo;
    // lower_bound(v+1)
    hi = nEdges;
    while (lo < hi) { int mid = (lo + hi) >> 1; if (src[mid] < v + 1) lo = mid + 1; else hi = mid; }
    int end = lo;

    float acc = 0.0f;
    for (int e = beg; e < end; ++e) {
        int d = dst[e];                                  // uniform per wave
        acc += x[(size_t)d * FEAT + f];                  // coalesced 128B per wave
    }
    agg[(size_t)v * FEAT + f] = acc;
    if (f == 0) deg[v] = (float)(end - beg);
}

// ---------------------------------------------------------------------------
// Kernel 2: out = [x | agg/deg] @ W^T + b, then row-L2-normalize.
// 8 waves/block, one 16-node tile per wave, f32 WMMA 16x16x4, K = 128.
// ---------------------------------------------------------------------------
__global__ __launch_bounds__(256) void sage_gemm_kernel(
    const float* __restrict__ x,
    const float* __restrict__ W,      // [64, 128] row-major
    const float* __restrict__ bias,   // [64]
    const float* __restrict__ agg,
    const float* __restrict__ deg,
    float* __restrict__ out,          // [N, 64]
    int nNodes)
{
    __shared__ float Wl[FEAT * WSTRIDE];   // 64 rows x 132 floats = 33.8 KB

    int tid = threadIdx.x;
    for (int i = tid; i < FEAT * 128; i += 256) {
        Wl[(i >> 7) * WSTRIDE + (i & 127)] = W[i];
    }
    __syncthreads();

    int lane = tid & 31;
    int wave = tid >> 5;
    int ln   = lane & 15;     // N index within tile / M index for A frag
    int kh   = lane >> 4;     // K-half select for A/B fragments

    int m0 = (blockIdx.x * 8 + wave) * 16;
    int m  = m0 + ln;
    int mc = (m < nNodes) ? m : (nNodes - 1);

    float dg = deg[mc];
    float rd = 1.0f / fmaxf(dg, 1.0f);

    const float* xrow = x   + (size_t)mc * FEAT;
    const float* arow = agg + (size_t)mc * FEAT;

    v8f acc0 = {}; v8f acc1 = {}; v8f acc2 = {}; v8f acc3 = {};

#pragma unroll
    for (int kk = 0; kk < 32; ++kk) {
        const int k0 = kk * 4;
        // A fragment: lane holds A[M=ln][k0+kh*2] and A[M=ln][k0+kh*2+1]
        v2f a;
        if (k0 < 64) {
            float2 t = *(const float2*)(xrow + k0 + kh * 2);
            a.x = t.x; a.y = t.y;
        } else {
            float2 t = *(const float2*)(arow + (k0 - 64) + kh * 2);
            a.x = t.x * rd; a.y = t.y * rd;
        }
        // B fragments: B[k][n] = W[n][k]; lane holds (K=k0+kh*2, N=n0+ln) pair
        const float* wb = &Wl[ln * WSTRIDE + k0 + kh * 2];
        v2f b0, b1, b2, b3;
        { float2 t = *(const float2*)(wb + 0 * 16 * WSTRIDE); b0.x = t.x; b0.y = t.y; }
        { float2 t = *(const float2*)(wb + 1 * 16 * WSTRIDE); b1.x = t.x; b1.y = t.y; }
        { float2 t = *(const float2*)(wb + 2 * 16 * WSTRIDE); b2.x = t.x; b2.y = t.y; }
        { float2 t = *(const float2*)(wb + 3 * 16 * WSTRIDE); b3.x = t.x; b3.y = t.y; }

        acc0 = __builtin_amdgcn_wmma_f32_16x16x4_f32(false, a, false, b0, (short)0, acc0, false, false);
        acc1 = __builtin_amdgcn_wmma_f32_16x16x4_f32(false, a, false, b1, (short)0, acc1, false, false);
        acc2 = __builtin_amdgcn_wmma_f32_16x16x4_f32(false, a, false, b2, (short)0, acc2, false, false);
        acc3 = __builtin_amdgcn_wmma_f32_16x16x4_f32(false, a, false, b3, (short)0, acc3, false, false);
    }

    // bias: depends only on output column N = nt*16 + ln
    float bb0 = bias[ 0 + ln];
    float bb1 = bias[16 + ln];
    float bb2 = bias[32 + ln];
    float bb3 = bias[48 + ln];
#pragma unroll
    for (int r = 0; r < 8; ++r) {
        acc0[r] += bb0; acc1[r] += bb1; acc2[r] += bb2; acc3[r] += bb3;
    }

    // Row-wise L2 norm + store.
    // C/D layout: VGPR r -> M = r (lanes 0-15), M = r+8 (lanes 16-31); N = ln.
#pragma unroll
    for (int r = 0; r < 8; ++r) {
        float s = acc0[r] * acc0[r] + acc1[r] * acc1[r]
                + acc2[r] * acc2[r] + acc3[r] * acc3[r];
        s += __shfl_xor(s, 1);
        s += __shfl_xor(s, 2);
        s += __shfl_xor(s, 4);
        s += __shfl_xor(s, 8);        // full 16-lane half reduced (wave32)
        float sc = 1.0f / fmaxf(sqrtf(s), 1e-12f);

        int node = m0 + r + kh * 8;
        if (node < nNodes) {
            size_t o = (size_t)node * FEAT + ln;
            out[o +  0] = acc0[r] * sc;
            out[o + 16] = acc1[r] * sc;
            out[o + 32] = acc2[r] * sc;
            out[o + 48] = acc3[r] * sc;
        }
    }
}

// ---------------------------------------------------------------------------
extern "C" void kernel_launch(void* const* d_in, const int* in_sizes, int n_in,
                              void* d_out, int out_size, void* d_ws, size_t ws_size,
                              hipStream_t stream) {
    const float* x  = (const float*)d_in[0];   // [N, 64]
    const float* W  = (const float*)d_in[1];   // [64, 128]
    const float* b  = (const float*)d_in[2];   // [64]
    const int*   ei = (const int*)d_in[3];     // [2, E]

    const int N = in_sizes[0] / FEAT;
    const int E = in_sizes[3] / 2;
    const int* src = ei;
    const int* dst = ei + E;

    // Workspace layout: deg [N floats, padded], then agg [N*64 floats]
    size_t degpad = (((size_t)N + 1023) / 1024) * 1024;
    float* deg = (float*)d_ws;
    float* agg = deg + degpad;
    float* out = (float*)d_out;

    // 1) deterministic aggregation (writes every agg/deg slot exactly once)
    {
        long long tot = (long long)N * 64;
        int blocks = (int)((tot + 255) / 256);
        sage_agg_kernel<<<blocks, 256, 0, stream>>>(x, src, dst, agg, deg, N, E);
    }
    // 2) WMMA GEMM + bias + L2 normalize
    {
        int tiles  = (N + 15) / 16;
        int blocks = (tiles + 7) / 8;
        sage_gemm_kernel<<<blocks, 256, 0, stream>>>(x, W, b, agg, deg, out, N);
    }
}